// OrientationExtractor_58119497449831
// MI455X (gfx1250) — compile-verified
//
#include <hip/hip_runtime.h>
#include <math.h>

// ---------------------------------------------------------------------------
// OrientationExtractor for MI455X (gfx1250, wave32)
//   k0: build 1024x64 lanczos3 interpolation matrix (shared by both axes)
//   k1: fused gray+grad+soft-bin+33x33 radial aggregation -> hog[8,9,64,64]
//   k2: per-batch max of cell bin-vector norm
//   k3: 2x2 eigen -> dominant orientation dom[8,2,64,64]
//   k4: width lanczos resize as GEMM via v_wmma_f32_16x16x4_f32
//   k5: height lanczos resize as GEMM via v_wmma_f32_16x16x4_f32 + fused
//       per-pixel 2-vector normalization epilogue -> d_out[8,2,1024,1024]
// ---------------------------------------------------------------------------

#define NBINS 9
#define CELLSZ 16
#define IMH 1024
#define IMW 1024
#define BATCH 8
#define HC 64          // cells per spatial dim
#define KWIN 33        // aggregation window
#define TILE 35        // gray tile (window + gradient halo)

typedef __attribute__((ext_vector_type(2))) float v2f;
typedef __attribute__((ext_vector_type(8))) float v8f;

__device__ __forceinline__ int reflect_idx(int q, int n) {
    // numpy 'reflect' (no edge repeat); single fold is enough for pad=16<<n
    if (q < 0) q = -q;
    if (q > n - 1) q = 2 * (n - 1) - q;
    return q;
}

// ------------------------- k0: lanczos weight matrix -----------------------
__global__ void build_lanczos_w(float* __restrict__ WL) {
    int i = blockIdx.x * blockDim.x + threadIdx.x;
    if (i >= IMH) return;
    const float PI = 3.14159265358979323846f;
    float s = (i + 0.5f) / 16.0f - 0.5f;   // source-space sample position
    float sum = 0.0f;
    for (int x = 0; x < HC; ++x) {
        float d = s - (float)x;
        float ad = fabsf(d);
        float w = 0.0f;
        if (ad < 3.0f) {
            if (ad < 1e-6f) w = 1.0f;
            else {
                float pd = PI * d, pd3 = pd * (1.0f / 3.0f);
                w = (sinf(pd) / pd) * (sinf(pd3) / pd3);
            }
        }
        sum += w;
    }
    float inv = 1.0f / sum;                // jax normalizes rows to sum 1
    for (int x = 0; x < HC; ++x) {
        float d = s - (float)x;
        float ad = fabsf(d);
        float w = 0.0f;
        if (ad < 3.0f) {
            if (ad < 1e-6f) w = 1.0f;
            else {
                float pd = PI * d, pd3 = pd * (1.0f / 3.0f);
                w = (sinf(pd) / pd) * (sinf(pd3) / pd3);
            }
        }
        WL[i * HC + x] = w * inv;
    }
}

// ------------------------- k1: fused HOG cell kernel -----------------------
// One 256-thread block per output cell (b,ci,cj). Gray staged in LDS over the
// original-coordinate span of the reflect-padded 33x33 window (+/-1 gradient
// halo with zero pad). Input x stays resident in the 192MB L2.
__global__ void hog_cells(const float* __restrict__ x, float* __restrict__ hog) {
    const int cj = blockIdx.x, ci = blockIdx.y, b = blockIdx.z;
    const int tid = threadIdx.x;
    __shared__ float tile[TILE][TILE + 1];
    __shared__ float red[NBINS][8];

    const int base_i = ci * CELLSZ - CELLSZ;
    const int base_j = cj * CELLSZ - CELLSZ;
    const int lo_i = max(0, base_i);
    const int lo_j = max(0, base_j);

    for (int idx = tid; idx < TILE * TILE; idx += 256) {
        int tr = idx / TILE, tc = idx % TILE;
        int oi = lo_i - 1 + tr;
        int oj = lo_j - 1 + tc;
        float g = 0.0f;                    // zero pad for gradient halo
        if (oi >= 0 && oi < IMH && oj >= 0 && oj < IMW) {
            size_t p = ((size_t)(b * 3) * IMH + oi) * IMW + oj;
            g = 0.299f * x[p] + 0.587f * x[p + (size_t)IMH * IMW]
              + 0.114f * x[p + 2 * (size_t)IMH * IMW];
        }
        tile[tr][tc] = g;
    }
    __syncthreads();

    float acc[NBINS];
#pragma unroll
    for (int k = 0; k < NBINS; ++k) acc[k] = 0.0f;

    const float PI = 3.14159265358979323846f;
    for (int p = tid; p < KWIN * KWIN; p += 256) {
        int pi = p / KWIN, pj = p % KWIN;
        int mi = reflect_idx(base_i + pi, IMH);
        int mj = reflect_idx(base_j + pj, IMW);
        int ti = mi - (lo_i - 1);
        int tj = mj - (lo_j - 1);
        float gx = tile[ti][tj + 1] - tile[ti][tj - 1];
        float gy = tile[ti + 1][tj] - tile[ti - 1][tj];
        float inten = sqrtf(gx * gx + gy * gy + 1e-12f);
        float ang = fmodf(atan2f(gy, gx) + PI, PI);
        float orient = ang * (180.0f / PI);
        float di = (float)(pi - CELLSZ), dj = (float)(pj - CELLSZ);
        float f = 1.0f - sqrtf(di * di + dj * dj) * (1.0f / (CELLSZ * 1.41421356237f));
        float wgt = inten * f;
#pragma unroll
        for (int k = 0; k < NBINS; ++k) {
            float c = 10.0f + 20.0f * (float)k;
            float wk = 1.0f - fabsf(orient - c) * (1.0f / 20.0f);
            wk = fminf(fmaxf(wk, 0.0f), 1.0f);
            acc[k] += wk * wgt;
        }
    }

    // wave32 butterfly reduce, then cross-wave via LDS
#pragma unroll
    for (int k = 0; k < NBINS; ++k)
        for (int off = 16; off > 0; off >>= 1)
            acc[k] += __shfl_xor(acc[k], off, 32);
    int wave = tid >> 5, lane = tid & 31;
    if (lane == 0) {
#pragma unroll
        for (int k = 0; k < NBINS; ++k) red[k][wave] = acc[k];
    }
    __syncthreads();
    if (tid < NBINS) {
        float s = 0.0f;
        for (int w = 0; w < 8; ++w) s += red[tid][w];
        hog[(((size_t)b * NBINS + tid) * HC + ci) * HC + cj] = s;
    }
}

// ------------------------- k2: per-batch max cell norm ---------------------
__global__ void batch_max(const float* __restrict__ hog, float* __restrict__ maxn) {
    int b = blockIdx.x, tid = threadIdx.x;
    __shared__ float red[8];
    float m = 0.0f;
    for (int cell = tid; cell < HC * HC; cell += 256) {
        float s = 0.0f;
#pragma unroll
        for (int k = 0; k < NBINS; ++k) {
            float h = hog[((size_t)b * NBINS + k) * (HC * HC) + cell];
            s += h * h;
        }
        m = fmaxf(m, s);
    }
    for (int off = 16; off > 0; off >>= 1)
        m = fmaxf(m, __shfl_xor(m, off, 32));
    if ((tid & 31) == 0) red[tid >> 5] = m;
    __syncthreads();
    if (tid == 0) {
        float mm = red[0];
        for (int w = 1; w < 8; ++w) mm = fmaxf(mm, red[w]);
        maxn[b] = sqrtf(mm);
    }
}

// ------------------------- k3: dominant orientation ------------------------
__global__ void dominant_orient(const float* __restrict__ hog,
                                const float* __restrict__ maxn,
                                float* __restrict__ dom) {
    int idx = blockIdx.x * blockDim.x + threadIdx.x;
    if (idx >= BATCH * HC * HC) return;
    int b = idx / (HC * HC);
    int cell = idx % (HC * HC);
    float inv = 1.0f / maxn[b];
    const float PI = 3.14159265358979323846f;
    float a = 0.0f, bb = 0.0f, d = 0.0f;
#pragma unroll
    for (int k = 0; k < NBINS; ++k) {
        float h = hog[((size_t)b * NBINS + k) * (HC * HC) + cell] * inv;
        float h2 = h * h;
        float th = (10.0f + 20.0f * (float)k) * (PI / 180.0f);
        float s = sinf(th), c = cosf(th);
        a += s * s * h2;
        bb += -s * c * h2;
        d += c * c * h2;
    }
    float half = 0.5f * (a - d);
    float disc = sqrtf(half * half + bb * bb + 1e-20f);
    float lam = 0.5f * (a + d) + disc;
    float v1x = bb, v1y = lam - a;
    float v2x = lam - d, v2y = bb;
    float n1 = v1x * v1x + v1y * v1y;
    float n2 = v2x * v2x + v2y * v2y;
    float vx = (n1 > n2) ? v1x : v2x;
    float vy = (n1 > n2) ? v1y : v2y;
    float nrm = 1.0f / sqrtf(vx * vx + vy * vy + 1e-9f);
    vx *= nrm; vy *= nrm;
    float sgn = (vx >= 0.0f) ? 1.0f : -1.0f;
    dom[(size_t)(b * 2 + 0) * (HC * HC) + cell] = vx * sgn;
    dom[(size_t)(b * 2 + 1) * (HC * HC) + cell] = vy * sgn;
}

// ------------------------- k4: width resize GEMM (WMMA) --------------------
// tmpW[1024 x 1024] = dom[1024 rows x 64] * WL^T[64 x 1024]. One wave per
// 16x16 tile; A/B fragments per CDNA5 16x16x4 f32 VGPR layout.
__global__ void resize_w_wmma(const float* __restrict__ dom,
                              const float* __restrict__ WL,
                              float* __restrict__ tmpW) {
    const int Nt = blockIdx.x, Mt = blockIdx.y;
    const int lane = threadIdx.x;
    const int half = lane >> 4;          // K-pair select (0 -> K0/1, 1 -> K2/3)
    const int l15 = lane & 15;
    const int koff = half * 2;
    v8f acc = {};
    const float* arow = dom + (size_t)(Mt * 16 + l15) * HC;     // A[M][k]
    const float* brow = WL + (size_t)(Nt * 16 + l15) * HC;      // B[k][N]=WL[N][k]
#pragma unroll
    for (int kk = 0; kk < 16; ++kk) {
        int k0 = kk * 4 + koff;
        v2f a, bf;
        a.x = arow[k0];  a.y = arow[k0 + 1];
        bf.x = brow[k0]; bf.y = brow[k0 + 1];
        acc = __builtin_amdgcn_wmma_f32_16x16x4_f32(false, a, false, bf,
                                                    (short)0, acc, false, false);
    }
#pragma unroll
    for (int r = 0; r < 8; ++r) {
        int row = Mt * 16 + r + 8 * half;
        int col = Nt * 16 + l15;
        tmpW[(size_t)row * IMW + col] = acc[r];
    }
}

// -------------- k5: height resize GEMM (WMMA) + normalize epilogue ---------
// out[b,c,Y,X] = sum_y WL[Y,y]*tmpW[b,c,y,X]; both channels in one wave so the
// 2-vector normalization happens on the accumulator fragments.
__global__ void resize_h_norm_wmma(const float* __restrict__ tmpW,
                                   const float* __restrict__ WL,
                                   float* __restrict__ out) {
    const int Xt = blockIdx.x, Yt = blockIdx.y, b = blockIdx.z;
    const int lane = threadIdx.x;
    const int half = lane >> 4;
    const int l15 = lane & 15;
    const int koff = half * 2;
    v8f acc0 = {}, acc1 = {};
    const float* arow = WL + (size_t)(Yt * 16 + l15) * HC;                 // A[M][k]
    const float* b0 = tmpW + (size_t)(b * 2 + 0) * HC * IMW + Xt * 16 + l15;
    const float* b1 = tmpW + (size_t)(b * 2 + 1) * HC * IMW + Xt * 16 + l15;
#pragma unroll
    for (int kk = 0; kk < 16; ++kk) {
        int k0 = kk * 4 + koff;
        v2f a, f0, f1;
        a.x = arow[k0]; a.y = arow[k0 + 1];
        f0.x = b0[(size_t)k0 * IMW]; f0.y = b0[(size_t)(k0 + 1) * IMW];
        f1.x = b1[(size_t)k0 * IMW]; f1.y = b1[(size_t)(k0 + 1) * IMW];
        acc0 = __builtin_amdgcn_wmma_f32_16x16x4_f32(false, a, false, f0,
                                                     (short)0, acc0, false, false);
        acc1 = __builtin_amdgcn_wmma_f32_16x16x4_f32(false, a, false, f1,
                                                     (short)0, acc1, false, false);
    }
#pragma unroll
    for (int r = 0; r < 8; ++r) {
        int Y = Yt * 16 + r + 8 * half;
        int X = Xt * 16 + l15;
        float v0 = acc0[r], v1 = acc1[r];
        float inv = 1.0f / sqrtf(v0 * v0 + v1 * v1 + 1e-9f);
        out[((size_t)(b * 2 + 0) * IMH + Y) * IMW + X] = v0 * inv;
        out[((size_t)(b * 2 + 1) * IMH + Y) * IMW + X] = v1 * inv;
    }
}

// ---------------------------------------------------------------------------
extern "C" void kernel_launch(void* const* d_in, const int* in_sizes, int n_in,
                              void* d_out, int out_size, void* d_ws, size_t ws_size,
                              hipStream_t stream) {
    (void)in_sizes; (void)n_in; (void)out_size; (void)ws_size;
    const float* x = (const float*)d_in[0];
    float* out = (float*)d_out;

    // workspace layout (floats): total ~1.47M floats (~5.9 MB)
    float* ws   = (float*)d_ws;
    float* WL   = ws;                                  // 1024*64      = 65536
    float* hog  = WL + IMH * HC;                       // 8*9*64*64    = 294912
    float* maxn = hog + (size_t)BATCH * NBINS * HC * HC; // 8
    float* dom  = maxn + BATCH;                        // 8*2*64*64    = 65536
    float* tmpW = dom + (size_t)BATCH * 2 * HC * HC;   // 8*2*64*1024  = 1048576

    build_lanczos_w<<<4, 256, 0, stream>>>(WL);
    hog_cells<<<dim3(HC, HC, BATCH), 256, 0, stream>>>(x, hog);
    batch_max<<<BATCH, 256, 0, stream>>>(hog, maxn);
    dominant_orient<<<(BATCH * HC * HC + 255) / 256, 256, 0, stream>>>(hog, maxn, dom);
    resize_w_wmma<<<dim3(IMW / 16, (BATCH * 2 * HC) / 16), 32, 0, stream>>>(dom, WL, tmpW);
    resize_h_norm_wmma<<<dim3(IMW / 16, IMH / 16, BATCH), 32, 0, stream>>>(tmpW, WL, out);
}